// XGNN_model_50500225466810
// MI455X (gfx1250) — compile-verified
//
#include <hip/hip_runtime.h>
#include <hip/hip_bf16.h>
#include <math.h>

typedef __attribute__((ext_vector_type(2))) float v2f;
typedef __attribute__((ext_vector_type(8))) float v8f;

#define F_OUT 16

// ---------------------------------------------------------------------------
// WMMA GEMM: out[n x 16] = X[n x K] @ W[K x 16]
// X rows [0, n0) come from X0, rows [n0, n) from X1 (concat of x and candidates).
// One wave32 per 16-row tile; V_WMMA_F32_16X16X4_F32, K streamed in steps of 4.
// A layout (16x4 f32): lanes 0-15 -> M=lane, K={0,1}; lanes 16-31 -> M=lane-16, K={2,3}
// B layout (4x16 f32): lanes 0-15 -> N=lane, K-rows {0,1}; lanes 16-31 -> K-rows {2,3}
// C/D layout: VGPR r -> (M=r, N=lane) for lanes 0-15, (M=r+8, N=lane-16) for 16-31
// ---------------------------------------------------------------------------
__global__ void gemm16_wmma(const float* __restrict__ X0,
                            const float* __restrict__ X1,
                            int n0, int n, int K,
                            const float* __restrict__ W,
                            float* __restrict__ out) {
    const int lane = threadIdx.x & 31;
    const int wave = threadIdx.x >> 5;
    const int tile = blockIdx.x * 8 + wave;
    const int ntiles = (n + 15) >> 4;
    if (tile >= ntiles) return;              // uniform per-wave exit -> EXEC all 1s

    const int half = lane >> 4;              // 0: K-pair {0,1}; 1: K-pair {2,3}
    const int l    = lane & 15;
    int m = tile * 16 + l;
    int mc = (m < n) ? m : (n - 1);          // clamp for ragged last tile
    const float* row = (mc < n0) ? (X0 + (size_t)mc * K)
                                 : (X1 + (size_t)(mc - n0) * K);

    v8f c = {};
    for (int kb = 0; kb < K; kb += 4) {
        v2f a, b;
        a.x = row[kb + 2 * half + 0];
        a.y = row[kb + 2 * half + 1];
        b.x = W[(kb + 2 * half + 0) * F_OUT + l];
        b.y = W[(kb + 2 * half + 1) * F_OUT + l];
        c = __builtin_amdgcn_wmma_f32_16x16x4_f32(false, a, false, b,
                                                  (short)0, c, false, false);
    }

    #pragma unroll
    for (int r = 0; r < 8; ++r) {
        int orow = tile * 16 + r + 8 * half;
        if (orow < n) out[(size_t)orow * F_OUT + l] = c[r];
    }
}

// ---------------------------------------------------------------------------
// Degree / normalization
// ---------------------------------------------------------------------------
__global__ void deg_init(float* __restrict__ deg, int n) {
    int i = blockIdx.x * blockDim.x + threadIdx.x;
    if (i < n) deg[i] = 1.0f;                // self-loop
}

__global__ void deg_count(const int* __restrict__ ei, float* __restrict__ deg, int nE) {
    int e = blockIdx.x * blockDim.x + threadIdx.x;
    if (e < nE) atomicAdd(&deg[ei[nE + e]], 1.0f);   // dst row of edge_index
}

__global__ void rsqrt_inplace(float* __restrict__ deg, int n) {
    int i = blockIdx.x * blockDim.x + threadIdx.x;
    if (i < n) deg[i] = rsqrtf(deg[i]);
}

__global__ void fzero4(float4* __restrict__ p, int nv) {
    int i = blockIdx.x * blockDim.x + threadIdx.x;
    if (i < nv) p[i] = make_float4(0.f, 0.f, 0.f, 0.f);
}

// ---------------------------------------------------------------------------
// Edge scatter: agg[dst] += h[src] * dinv[src]*dinv[dst].  4 lanes per edge,
// float4 gather + 4 global f32 atomics each.
// ---------------------------------------------------------------------------
__global__ void agg_scatter(const float* __restrict__ h,
                            const int* __restrict__ ei,
                            const float* __restrict__ dinv,
                            float* __restrict__ agg, int nE) {
    long long t = (long long)blockIdx.x * blockDim.x + threadIdx.x;
    int e = (int)(t >> 2);
    if (e >= nE) return;
    int q = (int)(t & 3);
    int src = ei[e];
    int dst = ei[nE + e];
    float coef = dinv[src] * dinv[dst];
    const float4 hv = *(const float4*)(h + (size_t)src * F_OUT + q * 4);
    float* base = agg + (size_t)dst * F_OUT + q * 4;
    atomicAdd(base + 0, hv.x * coef);
    atomicAdd(base + 1, hv.y * coef);
    atomicAdd(base + 2, hv.z * coef);
    atomicAdd(base + 3, hv.w * coef);
}

// out = relu(agg + pre * dinv^2 + bias)
__global__ void conv_finalize(const float* __restrict__ pre,
                              const float* __restrict__ agg,
                              const float* __restrict__ dinv,
                              const float* __restrict__ bias,
                              float* __restrict__ out, int n) {
    int i = blockIdx.x * blockDim.x + threadIdx.x;
    if (i >= n) return;
    float d2 = dinv[i] * dinv[i];
    #pragma unroll
    for (int j = 0; j < F_OUT; ++j) {
        float v = agg[(size_t)i * F_OUT + j] + pre[(size_t)i * F_OUT + j] * d2 + bias[j];
        out[(size_t)i * F_OUT + j] = fmaxf(v, 0.0f);
    }
}

// ---------------------------------------------------------------------------
// Per-node MLPs (tiny; VALU)
// ---------------------------------------------------------------------------
__global__ void mlp1_kernel(const float* __restrict__ feat,
                            const float* __restrict__ w1, const float* __restrict__ b1,
                            const float* __restrict__ w2, const float* __restrict__ b2,
                            float* __restrict__ out1, int n) {
    int i = blockIdx.x * blockDim.x + threadIdx.x;
    if (i >= n) return;
    float f[16];
    #pragma unroll
    for (int j = 0; j < 16; ++j) f[j] = feat[(size_t)i * 16 + j];
    float acc = b2[0];
    #pragma unroll
    for (int k = 0; k < 16; ++k) {
        float s = b1[k];
        #pragma unroll
        for (int j = 0; j < 16; ++j) s += f[j] * w1[j * 16 + k];
        acc += fmaxf(s, 0.0f) * w2[k];
    }
    out1[i] = acc;
}

__global__ void mlp2_kernel(const float* __restrict__ feat,
                            const int* __restrict__ startIdx,
                            const float* __restrict__ w1, const float* __restrict__ b1,
                            const float* __restrict__ w2, const float* __restrict__ b2,
                            float* __restrict__ out2, int n) {
    int i = blockIdx.x * blockDim.x + threadIdx.x;
    if (i >= n) return;
    int s0 = *startIdx;
    float f[32];
    #pragma unroll
    for (int j = 0; j < 16; ++j) f[j] = feat[(size_t)i * 16 + j];
    #pragma unroll
    for (int j = 0; j < 16; ++j) f[16 + j] = feat[(size_t)s0 * 16 + j];
    float acc = b2[0];
    for (int k = 0; k < 24; ++k) {
        float s = b1[k];
        #pragma unroll
        for (int j = 0; j < 32; ++j) s += f[j] * w1[j * 24 + k];
        acc += fmaxf(s, 0.0f) * w2[k];
    }
    out2[i] = acc;
}

// ---------------------------------------------------------------------------
// Argmax (first-occurrence tie-break, matching jnp.argmax), optional exclusion
// ---------------------------------------------------------------------------
__global__ void argmax_partial(const float* __restrict__ v, int n,
                               const int* __restrict__ exclPtr,
                               float* __restrict__ pval, int* __restrict__ pidx) {
    __shared__ float sv[256];
    __shared__ int   si[256];
    int tid = threadIdx.x;
    int excl = exclPtr ? *exclPtr : -1;
    float best = -__builtin_inff();
    int bi = 0x7fffffff;
    for (int i = blockIdx.x * 256 + tid; i < n; i += gridDim.x * 256) {
        if (i == excl) continue;
        float x = v[i];
        if (x > best || (x == best && i < bi)) { best = x; bi = i; }
    }
    sv[tid] = best; si[tid] = bi;
    __syncthreads();
    for (int s = 128; s > 0; s >>= 1) {
        if (tid < s) {
            if (sv[tid + s] > sv[tid] ||
                (sv[tid + s] == sv[tid] && si[tid + s] < si[tid])) {
                sv[tid] = sv[tid + s]; si[tid] = si[tid + s];
            }
        }
        __syncthreads();
    }
    if (tid == 0) { pval[blockIdx.x] = sv[0]; pidx[blockIdx.x] = si[0]; }
}

__global__ void argmax_final(const float* __restrict__ pval, const int* __restrict__ pidx,
                             int np, int* __restrict__ outIdx) {
    __shared__ float sv[256];
    __shared__ int   si[256];
    int tid = threadIdx.x;
    float best = -__builtin_inff();
    int bi = 0x7fffffff;
    for (int i = tid; i < np; i += 256) {
        float x = pval[i]; int ix = pidx[i];
        if (x > best || (x == best && ix < bi)) { best = x; bi = ix; }
    }
    sv[tid] = best; si[tid] = bi;
    __syncthreads();
    for (int s = 128; s > 0; s >>= 1) {
        if (tid < s) {
            if (sv[tid + s] > sv[tid] ||
                (sv[tid + s] == sv[tid] && si[tid + s] < si[tid])) {
                sv[tid] = sv[tid + s]; si[tid] = si[tid + s];
            }
        }
        __syncthreads();
    }
    if (tid == 0) outIdx[0] = si[0];
}

// d_out head: [0]=start, [1]=end, [2]=is_new_node, [3..130]=xall[end] (128 f32)
__global__ void write_head(const int* __restrict__ startIdx, const int* __restrict__ endIdx,
                           const float* __restrict__ x, const float* __restrict__ cand,
                           int n_data, float* __restrict__ out) {
    int j = threadIdx.x;   // 128 threads
    int e = *endIdx;
    const float* row = (e < n_data) ? (x + (size_t)e * 128)
                                    : (cand + (size_t)(e - n_data) * 128);
    out[3 + j] = row[j];
    if (j == 0) {
        out[0] = (float)(*startIdx);
        out[1] = (float)e;
        out[2] = (e >= n_data) ? 1.0f : 0.0f;
    }
}

// ---------------------------------------------------------------------------
extern "C" void kernel_launch(void* const* d_in, const int* in_sizes, int n_in,
                              void* d_out, int out_size, void* d_ws, size_t ws_size,
                              hipStream_t stream) {
    const float* x     = (const float*)d_in[0];
    const float* cand  = (const float*)d_in[1];
    const int*   ei    = (const int*)d_in[2];
    const float* cw1   = (const float*)d_in[3];
    const float* cb1   = (const float*)d_in[4];
    const float* cw2   = (const float*)d_in[5];
    const float* cb2   = (const float*)d_in[6];
    const float* m1w1  = (const float*)d_in[7];
    const float* m1b1  = (const float*)d_in[8];
    const float* m1w2  = (const float*)d_in[9];
    const float* m1b2  = (const float*)d_in[10];
    const float* m2w1  = (const float*)d_in[11];
    const float* m2b1  = (const float*)d_in[12];
    const float* m2w2  = (const float*)d_in[13];
    const float* m2b2  = (const float*)d_in[14];

    const int n_data = in_sizes[0] / 128;
    const int n_cand = in_sizes[1] / 128;
    const int nE     = in_sizes[2] / 2;
    const int n      = n_data + n_cand;

    // ---- workspace layout ----
    char* ws = (char*)d_ws;
    size_t off = 0;
    auto walloc = [&](size_t bytes) -> void* {
        void* p = ws + off;
        off = (off + bytes + 255) & ~(size_t)255;
        return p;
    };
    float* bufA = (float*)walloc((size_t)n * 16 * 4);   // pre-aggregation h
    float* bufB = (float*)walloc((size_t)n * 16 * 4);   // scatter accumulator
    float* bufC = (float*)walloc((size_t)n * 16 * 4);   // relu'd features
    float* deg  = (float*)walloc((size_t)n * 4);        // deg -> dinv in place
    float* pval = (float*)walloc(256 * 4);
    int*   pidx = (int*)walloc(256 * 4);
    int*   sIdx = (int*)walloc(4);
    int*   eIdx = (int*)walloc(4);

    float* out_f = (float*)d_out;
    float* out1  = out_f + 131;         // [n] raw logits
    float* out2  = out_f + 131 + n;     // [n] raw logits

    const int ntiles     = (n + 15) >> 4;
    const int gemmBlocks = (ntiles + 7) / 8;
    const int nThreads   = (n + 255) / 256;
    const int nv16       = n * 4;                       // float4 count of n*16
    const int zBlocks    = (nv16 + 255) / 256;
    const int eBlocks    = (nE + 255) / 256;
    const int sBlocks    = (int)(((long long)nE * 4 + 255) / 256);

    // GCN layer 1: hpre = xall @ W1  (WMMA, K=128)
    gemm16_wmma<<<gemmBlocks, 256, 0, stream>>>(x, cand, n_data, n, 128, cw1, bufA);

    // degree + dinv (shared by both conv layers)
    deg_init<<<nThreads, 256, 0, stream>>>(deg, n);
    deg_count<<<eBlocks, 256, 0, stream>>>(ei, deg, nE);
    rsqrt_inplace<<<nThreads, 256, 0, stream>>>(deg, n);

    // aggregate + finalize layer 1
    fzero4<<<zBlocks, 256, 0, stream>>>((float4*)bufB, nv16);
    agg_scatter<<<sBlocks, 256, 0, stream>>>(bufA, ei, deg, bufB, nE);
    conv_finalize<<<nThreads, 256, 0, stream>>>(bufA, bufB, deg, cb1, bufC, n);

    // GCN layer 2: hpre2 = h1 @ W2  (WMMA, K=16)
    gemm16_wmma<<<gemmBlocks, 256, 0, stream>>>(bufC, bufC, n, n, 16, cw2, bufA);
    fzero4<<<zBlocks, 256, 0, stream>>>((float4*)bufB, nv16);
    agg_scatter<<<sBlocks, 256, 0, stream>>>(bufA, ei, deg, bufB, nE);
    conv_finalize<<<nThreads, 256, 0, stream>>>(bufA, bufB, deg, cb2, bufC, n);

    // MLP1 logits -> d_out slot; masked argmax over data nodes (softmax is monotonic)
    mlp1_kernel<<<nThreads, 256, 0, stream>>>(bufC, m1w1, m1b1, m1w2, m1b2, out1, n);
    argmax_partial<<<256, 256, 0, stream>>>(out1, n_data, (const int*)nullptr, pval, pidx);
    argmax_final<<<1, 256, 0, stream>>>(pval, pidx, 256, sIdx);

    // MLP2 logits (concat features with starting node's features); argmax excl. start
    mlp2_kernel<<<nThreads, 256, 0, stream>>>(bufC, sIdx, m2w1, m2b1, m2w2, m2b2, out2, n);
    argmax_partial<<<256, 256, 0, stream>>>(out2, n, sIdx, pval, pidx);
    argmax_final<<<1, 256, 0, stream>>>(pval, pidx, 256, eIdx);

    // head: indices, flag, new_features = xall[ending_idx]
    write_head<<<1, 128, 0, stream>>>(sIdx, eIdx, x, cand, n_data, out_f);
}